// HistogramLoss_53180285059398
// MI455X (gfx1250) — compile-verified
//
#include <hip/hip_runtime.h>
#include <hip/hip_bf16.h>

// ---------------------------------------------------------------------------
// HistogramLoss for MI455X (gfx1250).
//   K0 zero-init scratch accumulators
//   K1 WMMA f32 GEMM: rot[p][d] = sum_k ortho[d][k] * feature[k][p]
//   K2 replica weights w[j][c] (4x4 label block histogram per feature pixel)
//   K3 per-class moments via LDS accumulation (counts, sum, sum^2)
//   K4 params miu / 1/std per (c,d)
//   K5 13-bin Gaussian histogram (exp-bound), TDM double-buffered LDS staging
//   K6 finalize: normalize, Huber vs constant target, masked mean -> loss
// ---------------------------------------------------------------------------

typedef __attribute__((ext_vector_type(2))) float v2f;
typedef __attribute__((ext_vector_type(8))) float v8f;
typedef unsigned int u32;
typedef __attribute__((ext_vector_type(4))) u32 u32x4;
typedef __attribute__((ext_vector_type(8))) int i32x8;
typedef __attribute__((ext_vector_type(4))) int i32x4;

#define D_DIM   256
#define P_DIM   16384      // 128*128 unique feature pixels
#define LBL_W   512
#define NCLS    19
#define NBINS   13

#if defined(__gfx1250__) && __has_builtin(__builtin_amdgcn_tensor_load_to_lds) && \
    __has_builtin(__builtin_amdgcn_s_wait_tensorcnt)
#define USE_TDM 1
#else
#define USE_TDM 0
#endif

// ---- workspace layout (floats) --------------------------------------------
#define ROT_OFF   0                        // 16384*256
#define W_OFF     (ROT_OFF + P_DIM*D_DIM)  // 16384*19
#define SUM1_OFF  (W_OFF + P_DIM*NCLS)     // 19*256
#define SUM2_OFF  (SUM1_OFF + NCLS*D_DIM)  // 19*256
#define CNT_OFF   (SUM2_OFF + NCLS*D_DIM)  // 32 (padded)
#define SV_OFF    (CNT_OFF + 32)           // 19*256*13
#define MIU_OFF   (SV_OFF + NCLS*D_DIM*NBINS)
#define RSTD_OFF  (MIU_OFF + NCLS*D_DIM)
#define ZERO_N    (SV_OFF + NCLS*D_DIM*NBINS - SUM1_OFF)   // sum1..sv contiguous

// K0 --------------------------------------------------------------------------
__global__ void k_zero(float* __restrict__ p, int n) {
    int i = blockIdx.x * 256 + threadIdx.x;
    if (i < n) p[i] = 0.0f;
}

// K1: WMMA GEMM  rot[p][d] = sum_k Q[d][k] * F[k][p] ---------------------------
__global__ void k_gemm(const float* __restrict__ Q,   // [256][256] row-major
                       const float* __restrict__ F,   // [256][16384]
                       float* __restrict__ rot) {     // [16384][256]
    const int wave = threadIdx.x >> 5;
    const int lane = threadIdx.x & 31;
    const int half = lane >> 4;          // ISA: lanes 16-31 hold the upper half
    const int l    = lane & 15;
    const int tile  = blockIdx.x * 8 + wave;   // 0..16383
    const int dtile = tile & 15;               // 16 d-tiles
    const int ptile = tile >> 4;               // 1024 p-tiles
    const int m = dtile * 16 + l;              // A row (output d)
    const int n = ptile * 16 + l;              // B col (pixel)

    const float* __restrict__ qrow = Q + (size_t)m * D_DIM;
    v8f c = {};
    for (int k0 = 0; k0 < D_DIM; k0 += 4) {
        const int ka = k0 + 2 * half;          // 16x4 f32 A: V0={K0,K2}, V1={K1,K3}
        v2f a, b;
        a.x = qrow[ka];
        a.y = qrow[ka + 1];
        b.x = F[(size_t)ka * P_DIM + n];       // 4x16 f32 B: same half-split over K
        b.y = F[(size_t)(ka + 1) * P_DIM + n];
        c = __builtin_amdgcn_wmma_f32_16x16x4_f32(
                /*neg_a=*/false, a, /*neg_b=*/false, b,
                /*c_mod=*/(short)0, c, /*reuse_a=*/false, /*reuse_b=*/false);
    }
    // C layout: VGPR v -> M = v + 8*half, N = l.  Store transposed: rot[p][d].
    float* dst = rot + (size_t)(ptile * 16 + l) * D_DIM + dtile * 16 + 8 * half;
    ((float4*)dst)[0] = make_float4(c[0], c[1], c[2], c[3]);
    ((float4*)dst)[1] = make_float4(c[4], c[5], c[6], c[7]);
}

// K2: replica weights ---------------------------------------------------------
__global__ void k_weights(const int* __restrict__ lab,   // [512][512]
                          float* __restrict__ w) {       // [16384][19]
    const int j  = blockIdx.x * 256 + threadIdx.x;       // 64 blocks
    const int py = j >> 7, px = j & 127;
    int v[16];
    #pragma unroll
    for (int dy = 0; dy < 4; ++dy)
        #pragma unroll
        for (int dx = 0; dx < 4; ++dx)
            v[dy * 4 + dx] = lab[(size_t)(py * 4 + dy) * LBL_W + px * 4 + dx];
    #pragma unroll
    for (int c = 0; c < NCLS; ++c) {
        int s = 0;
        #pragma unroll
        for (int i = 0; i < 16; ++i) s += (v[i] == c);
        w[(size_t)j * NCLS + c] = (float)s;
    }
}

// K3: per-class moments -------------------------------------------------------
__global__ void k_moments(const float* __restrict__ rot,
                          const float* __restrict__ w,
                          float* __restrict__ sum1,     // [19][256]
                          float* __restrict__ sum2,     // [19][256]
                          float* __restrict__ counts) { // [19]
    __shared__ float s1[NCLS * D_DIM];
    __shared__ float s2[NCLS * D_DIM];
    __shared__ float wrow[NCLS];
    __shared__ float cw[NCLS];
    const int t = threadIdx.x;
    for (int c = 0; c < NCLS; ++c) { s1[c * D_DIM + t] = 0.f; s2[c * D_DIM + t] = 0.f; }
    if (t < NCLS) cw[t] = 0.f;
    __syncthreads();

    const int j0 = blockIdx.x * 256;                    // 64 blocks x 256 pixels
    for (int jj = 0; jj < 256; ++jj) {
        const int j = j0 + jj;
        __builtin_prefetch(&rot[(size_t)(j + 2) * D_DIM + t], 0, 0); // global_prefetch_b8
        if (t < NCLS) wrow[t] = w[(size_t)j * NCLS + t];
        __syncthreads();
        const float x  = rot[(size_t)j * D_DIM + t];
        const float x2 = x * x;
        if (t < NCLS) cw[t] += wrow[t];
        for (int c = 0; c < NCLS; ++c) {
            const float wv = wrow[c];                   // uniform -> scalar branch
            if (wv != 0.f) {
                s1[c * D_DIM + t] = fmaf(wv, x,  s1[c * D_DIM + t]);
                s2[c * D_DIM + t] = fmaf(wv, x2, s2[c * D_DIM + t]);
            }
        }
        __syncthreads();
    }
    for (int c = 0; c < NCLS; ++c) {
        atomicAdd(&sum1[c * D_DIM + t], s1[c * D_DIM + t]);
        atomicAdd(&sum2[c * D_DIM + t], s2[c * D_DIM + t]);
    }
    if (t < NCLS) atomicAdd(&counts[t], cw[t]);
}

// K4: per-(c,d) params --------------------------------------------------------
__global__ void k_params(const float* __restrict__ sum1,
                         const float* __restrict__ sum2,
                         const float* __restrict__ counts,
                         float* __restrict__ miu,
                         float* __restrict__ rstd) {
    const int i = blockIdx.x * 256 + threadIdx.x;       // 19 blocks x 256
    const int c = i >> 8;
    const float n   = fmaxf(counts[c], 1.0f);
    const float m   = sum1[i] / n;
    const float var = sum2[i] / n - m * m + 1e-12f;
    miu[i]  = m;
    rstd[i] = 1.0f / sqrtf(var);
}

#if USE_TDM
// Issue one TDM 2D tile load: 16 rows x 256 f32 (16 KB) of rot[] -> LDS.
// D# per CDNA5 ISA 8.3/8.4: group0 = {flags, lds_addr, global_addr, type=2},
// group1 = {data_size=4B, tensor_dim0=256, tensor_dim1=16384, tile=256x16,
//           tensor_dim0_stride=256}.  Groups 2/3 zero (2D tensor).
// This toolchain exposes the 6-arg builtin form:
//   (u32x4 g0, i32x8 g1, i32x4 g2, i32x4 g3, i32x8 extra, i32 cpol)
__device__ __forceinline__ void tdm_load_tile(const float* gsrc, u32 lds_off) {
    const unsigned long long ga = (unsigned long long)(uintptr_t)gsrc;
    u32x4 g0;
    g0.x = 1u;                                          // count=1 (valid), load
    g0.y = lds_off;                                     // lds_addr [63:32]
    g0.z = (u32)(ga & 0xFFFFFFFFu);                     // global_addr [95:64]
    g0.w = (u32)((ga >> 32) & 0x01FFFFFFu) | (2u << 30);// addr[56:32] | type=2
    i32x8 g1;
    g1[0] = (int)(2u << 16);                            // data_size = 4 bytes
    g1[1] = (int)(256u << 16);                          // tensor_dim0 lo16 = 256
    g1[2] = (int)(16384u << 16);                        // dim0 hi=0 | tensor_dim1 lo16
    g1[3] = (int)(256u << 16);                          // dim1 hi=0 | tile_dim0 = 256
    g1[4] = 16;                                         // tile_dim1=16, tile_dim2=0
    g1[5] = 256;                                        // tensor_dim0_stride lo32
    g1[6] = 0;                                          // stride hi | dim1_stride lo
    g1[7] = 0;
    i32x4 gz4 = {0, 0, 0, 0};
    i32x8 gz8 = {0, 0, 0, 0, 0, 0, 0, 0};
    __builtin_amdgcn_tensor_load_to_lds(g0, g1, gz4, gz4, gz8, 0);
}
#endif

// K5: 13-bin Gaussian kernel histogram (the exp-bound hot loop) --------------
// exponent: -0.5*(delta + k*std)^2 * (25/var) == -12.5*((miu-x)/std + k)^2
// coef cancels under the later k-normalization, so it is omitted here.
__global__ void k_hist(const float* __restrict__ rot,
                       const float* __restrict__ w,
                       const float* __restrict__ miu,
                       const float* __restrict__ rstd,
                       float* __restrict__ sv) {        // [19][256][13]
    const int c = blockIdx.x;        // 19
    const int t = threadIdx.x;       // d
    const float mu = miu[c * D_DIM + t];
    const float rs = rstd[c * D_DIM + t];
    float acc[NBINS];
    #pragma unroll
    for (int k = 0; k < NBINS; ++k) acc[k] = 0.f;

    const int j0 = blockIdx.y * 256;                    // 64 pixel-chunks

#if USE_TDM
    __shared__ float buf[2][16][D_DIM];                 // 32 KB double buffer
    const bool issuer = (threadIdx.x < 32);             // wave 0 drives the TDM
    const u32 lds_base = (u32)(uintptr_t)&buf[0][0][0];
    if (issuer) tdm_load_tile(rot + (size_t)j0 * D_DIM, lds_base);
    for (int tile = 0; tile < 16; ++tile) {
        if (issuer) {
            if (tile < 15) {
                tdm_load_tile(rot + (size_t)(j0 + (tile + 1) * 16) * D_DIM,
                              lds_base + (u32)(((tile + 1) & 1) * 16 * D_DIM * 4));
                __builtin_amdgcn_s_wait_tensorcnt(1);   // oldest tile landed
            } else {
                __builtin_amdgcn_s_wait_tensorcnt(0);
            }
        }
        __syncthreads();                                // tile visible to all waves
        const float* bt = &buf[tile & 1][0][0];
        for (int r = 0; r < 16; ++r) {
            const int j = j0 + tile * 16 + r;
            const float wv = w[(size_t)j * NCLS + c];   // uniform -> s_load+s_cbranch
            if (wv != 0.f) {
                const float x = bt[r * D_DIM + t];
                float u = (mu - x) * rs - 3.0f;         // k = -3.0 start
                #pragma unroll
                for (int k = 0; k < NBINS; ++k) {
                    acc[k] = fmaf(wv, __expf(-12.5f * u * u), acc[k]);
                    u += 0.5f;
                }
            }
        }
        __syncthreads();                                // before buffer reuse
    }
#else
    for (int jj = 0; jj < 256; ++jj) {
        const int j = j0 + jj;
        const float wv = w[(size_t)j * NCLS + c];
        if (wv != 0.f) {
            const float x = rot[(size_t)j * D_DIM + t];
            float u = (mu - x) * rs - 3.0f;
            #pragma unroll
            for (int k = 0; k < NBINS; ++k) {
                acc[k] = fmaf(wv, __expf(-12.5f * u * u), acc[k]);
                u += 0.5f;
            }
        }
    }
#endif

    #pragma unroll
    for (int k = 0; k < NBINS; ++k)
        atomicAdd(&sv[((size_t)c * D_DIM + t) * NBINS + k], acc[k]);
}

// K6: finalize ----------------------------------------------------------------
__global__ void k_finalize(const float* __restrict__ sv,
                           const float* __restrict__ counts,
                           float* __restrict__ out) {
    __shared__ float red[256];
    const int t = threadIdx.x;
    // constant target: exp(-0.5 k^2)/Z   (the 1/sqrt(2*pi*var) factor cancels)
    float tgt[NBINS]; float Z = 0.f;
    #pragma unroll
    for (int k = 0; k < NBINS; ++k) {
        const float kk = -3.0f + 0.5f * k;
        tgt[k] = __expf(-0.5f * kk * kk);
        Z += tgt[k];
    }
    #pragma unroll
    for (int k = 0; k < NBINS; ++k) tgt[k] /= Z;

    float loss = 0.f;
    for (int c = 0; c < NCLS; ++c) {
        float s[NBINS]; float denom = 0.f;
        #pragma unroll
        for (int k = 0; k < NBINS; ++k) {
            s[k] = sv[((size_t)c * D_DIM + t) * NBINS + k];
            denom += s[k];
        }
        const float dn = (denom > 0.f) ? denom : 1.f;
        float h = 0.f;
        #pragma unroll
        for (int k = 0; k < NBINS; ++k) {
            const float diff = s[k] / dn - tgt[k];
            const float ad = fabsf(diff);
            h += (ad < 1.0f) ? 0.5f * diff * diff : ad - 0.5f;   // smooth_l1, beta=1
        }
        red[t] = h;
        __syncthreads();
        for (int off = 128; off > 0; off >>= 1) {
            if (t < off) red[t] += red[t + off];
            __syncthreads();
        }
        if (t == 0) {
            const float lc = red[0] / (float)(D_DIM * NBINS);
            if (c > 0 && counts[c] >= 1000.f) loss += lc;
        }
        __syncthreads();
    }
    if (t == 0) {
        float active = 0.f;
        for (int c = 0; c < NCLS; ++c) active += (counts[c] > 0.f) ? 1.f : 0.f;
        active = fmaxf(active, 1.0f);
        out[0] = 1.0f * loss / active;   // LOSS_WEIGHT = 1.0
    }
}

// ---------------------------------------------------------------------------
extern "C" void kernel_launch(void* const* d_in, const int* in_sizes, int n_in,
                              void* d_out, int out_size, void* d_ws, size_t ws_size,
                              hipStream_t stream) {
    const float* feature = (const float*)d_in[0];  // [1,256,128,128]
    const int*   label   = (const int*)d_in[1];    // [1,512,512]
    const float* ortho   = (const float*)d_in[2];  // [256,256]
    float* out = (float*)d_out;
    float* ws  = (float*)d_ws;

    float* rot    = ws + ROT_OFF;
    float* w      = ws + W_OFF;
    float* sum1   = ws + SUM1_OFF;
    float* sum2   = ws + SUM2_OFF;
    float* counts = ws + CNT_OFF;
    float* sv     = ws + SV_OFF;
    float* miu    = ws + MIU_OFF;
    float* rstd   = ws + RSTD_OFF;

    k_zero<<<(ZERO_N + 255) / 256, 256, 0, stream>>>(ws + SUM1_OFF, ZERO_N);
    k_gemm<<<2048, 256, 0, stream>>>(ortho, feature, rot);
    k_weights<<<P_DIM / 256, 256, 0, stream>>>(label, w);
    k_moments<<<64, 256, 0, stream>>>(rot, w, sum1, sum2, counts);
    k_params<<<NCLS, 256, 0, stream>>>(sum1, sum2, counts, miu, rstd);
    k_hist<<<dim3(NCLS, 64), 256, 0, stream>>>(rot, w, miu, rstd, sv);
    k_finalize<<<1, 256, 0, stream>>>(sv, counts, out);
}